// AttentionMLP_31834297598223
// MI455X (gfx1250) — compile-verified
//
#include <hip/hip_runtime.h>
#include <hip/hip_bf16.h>
#include <math.h>

// Problem constants (from reference): B=2, K=128, L=4096, DK=DL=HID=128
#define HID    128
#define KA     128
#define LB     4096
#define BATCH  2

typedef float  v2f_t  __attribute__((ext_vector_type(2)));
typedef float  v8f_t  __attribute__((ext_vector_type(8)));

// ---------------------------------------------------------------------------
// GEMM: Y[M x 128] = X[M x 128] @ W[128 x 128] (+ bias), fp32 WMMA 16x16x4.
// One wave per 16x16 output tile; grid sized exactly so EXEC is all-ones.
// (f32 WMMA path confirmed lowering to v_wmma_f32_16x16x4_f32 in round 1.)
// ---------------------------------------------------------------------------
__global__ __launch_bounds__(256) void gemm128_wmma(
    const float* __restrict__ X, const float* __restrict__ W,
    const float* __restrict__ bias, float* __restrict__ Y, int M) {
  const int lane   = threadIdx.x & 31;
  const int waveId = blockIdx.x * 8 + (threadIdx.x >> 5);
  const int tileM  = waveId >> 3;      // 128/16 = 8 N-tiles
  const int tileN  = waveId & 7;
  if (tileM * 16 >= M) return;         // wave-uniform exit

  const int half = lane >> 4;          // 0: lanes 0-15, 1: lanes 16-31
  const int l16  = lane & 15;
  const int col  = tileN * 16 + l16;   // N index this lane owns for B/C/D
  const float* Xr = X + (size_t)(tileM * 16 + l16) * HID;  // A row M=l16

  v8f_t acc = {};
  // A 16x4 f32 (ISA 7.12.2): lane<16 -> {K=0,K=1}, lane>=16 -> {K=2,K=3}
  // B 4x16 f32 (mirror):     v0 = row K=0|2 striped over N, v1 = row K=1|3
  #pragma unroll
  for (int k = 0; k < HID; k += 4) {
    const int kk = k + 2 * half;
    v2f_t a, b;
    a.x = Xr[kk];
    a.y = Xr[kk + 1];
    b.x = W[(size_t)kk * HID + col];
    b.y = W[(size_t)(kk + 1) * HID + col];
    acc = __builtin_amdgcn_wmma_f32_16x16x4_f32(
        false, a, false, b, (short)0, acc, false, false);
  }

  // C/D 16x16 f32: vgpr i -> row M = i + 8*half, col N = l16
  const float bv = bias ? bias[col] : 0.0f;
  #pragma unroll
  for (int i = 0; i < 8; ++i) {
    const int row = tileM * 16 + i + 8 * half;
    Y[(size_t)row * HID + col] = acc[i] + bv;
  }
}

// ---------------------------------------------------------------------------
// Raw scores, k-tiled: block = (16 k-rows) x (256 l's). 16 hA rows + w2 live
// in LDS; every sA/sW read is wave-uniform => LDS broadcast, no conflicts.
// Each thread streams one hB row once and emits 16 scores => hB is read
// K/16 = 8 times total (32 MB of L2 traffic) instead of 128 times (512 MB).
// scores[r, l] = sum_h relu(hA[r,h] + hB[b,l,h]) * w2[h]   (b1 folded in hA)
// ---------------------------------------------------------------------------
__global__ __launch_bounds__(256) void score_tiled(
    const float* __restrict__ hAb,  // (B*K, 128), bias folded in
    const float* __restrict__ hB,   // (B*L, 128)
    const float* __restrict__ w2,   // (128)
    float* __restrict__ out) {      // (B*K, L) raw scores
  const int tid = threadIdx.x;
  const int l   = blockIdx.x * 256 + tid;   // gridDim.x = L/256 = 16
  const int r0  = blockIdx.y * 16;          // gridDim.y = B*K/16 = 16
  const int b   = r0 >> 7;                  // K = 128

  __shared__ float sA[16][HID];
  __shared__ float sW[HID];

  for (int idx = tid; idx < 16 * HID; idx += 256)
    sA[idx >> 7][idx & (HID - 1)] =
        hAb[(size_t)(r0 + (idx >> 7)) * HID + (idx & (HID - 1))];
  if (tid < HID) sW[tid] = w2[tid];
  __syncthreads();

  const float4* rB = (const float4*)(hB + ((size_t)b * LB + l) * HID);

  float acc[16];
  #pragma unroll
  for (int k = 0; k < 16; ++k) acc[k] = 0.0f;

  #pragma unroll 4
  for (int q = 0; q < HID / 4; ++q) {
    const float4 v = rB[q];                       // per-lane global (L2-hot)
    const float4 w = ((const float4*)sW)[q];      // LDS broadcast
    #pragma unroll
    for (int k = 0; k < 16; ++k) {
      const float4 a = ((const float4*)sA[k])[q]; // LDS broadcast
      acc[k] += fmaxf(v.x + a.x, 0.0f) * w.x
              + fmaxf(v.y + a.y, 0.0f) * w.y
              + fmaxf(v.z + a.z, 0.0f) * w.z
              + fmaxf(v.w + a.w, 0.0f) * w.w;
    }
  }

  #pragma unroll
  for (int k = 0; k < 16; ++k)                    // coalesced: tid -> l
    out[(size_t)(r0 + k) * LB + l] = acc[k];
}

// ---------------------------------------------------------------------------
// In-place row softmax over L=4096. One 256-thread block (8 wave32) per row.
// ---------------------------------------------------------------------------
__global__ __launch_bounds__(256) void softmax_rows(float* __restrict__ out) {
  const int row = blockIdx.x;
  const int tid = threadIdx.x;
  __shared__ float red[8];

  float* o = out + (size_t)row * LB;
  float s[16];
  float m = -3.402823466e38f;
  #pragma unroll
  for (int i = 0; i < 16; ++i) {
    s[i] = o[i * 256 + tid];
    m = fmaxf(m, s[i]);
  }

  #pragma unroll
  for (int off = 16; off; off >>= 1) m = fmaxf(m, __shfl_xor(m, off, 32));
  if ((tid & 31) == 0) red[tid >> 5] = m;
  __syncthreads();
  float bm = red[0];
  #pragma unroll
  for (int i = 1; i < 8; ++i) bm = fmaxf(bm, red[i]);

  float sum = 0.0f;
  #pragma unroll
  for (int i = 0; i < 16; ++i) {
    const float e = __expf(s[i] - bm);
    s[i] = e;
    sum += e;
  }
  #pragma unroll
  for (int off = 16; off; off >>= 1) sum += __shfl_xor(sum, off, 32);
  __syncthreads();                    // red[] reads above complete
  if ((tid & 31) == 0) red[tid >> 5] = sum;
  __syncthreads();
  float bs = 0.0f;
  #pragma unroll
  for (int i = 0; i < 8; ++i) bs += red[i];
  const float inv = 1.0f / bs;

  #pragma unroll
  for (int i = 0; i < 16; ++i)
    o[i * 256 + tid] = s[i] * inv;
}

// ---------------------------------------------------------------------------
// inputs: [0]=inputA (2,128,128) f32, [1]=inputB (2,4096,128) f32,
//         [2]=W1 (256,128) f32, [3]=b1 (128) f32, [4]=w2 (128) f32,
//         [5]=nA, [6]=nB (scalars, ignored — shapes fixed).
// ws: hB = 8192*128 f32 (4 MB) @0; hA+b1 = 256*128 f32 (128 KB) after.
// ---------------------------------------------------------------------------
extern "C" void kernel_launch(void* const* d_in, const int* in_sizes, int n_in,
                              void* d_out, int out_size, void* d_ws, size_t ws_size,
                              hipStream_t stream) {
  (void)in_sizes; (void)n_in; (void)out_size; (void)ws_size;
  const float* A  = (const float*)d_in[0];
  const float* Bi = (const float*)d_in[1];
  const float* W1 = (const float*)d_in[2];
  const float* b1 = (const float*)d_in[3];
  const float* w2 = (const float*)d_in[4];
  float* out = (float*)d_out;

  float* hB  = (float*)d_ws;                                         // (B*L,128)
  float* hAb = (float*)((char*)d_ws + (size_t)BATCH * LB * HID * 4); // (B*K,128)

  // hB = inputB @ W1[128:]  -> 4096 tiles, 8 waves/block
  gemm128_wmma<<<512, 256, 0, stream>>>(Bi, W1 + (size_t)HID * HID, nullptr,
                                        hB, BATCH * LB);
  // hA+b1 = inputA @ W1[:128] + b1 -> 128 tiles
  gemm128_wmma<<<16, 256, 0, stream>>>(A, W1, b1, hAb, BATCH * KA);

  // raw scores into d_out: 16 (l-chunks) x 16 (k-tiles) blocks
  score_tiled<<<dim3(16, 16), 256, 0, stream>>>(hAb, hB, w2, out);

  // in-place softmax per (b,k) row
  softmax_rows<<<BATCH * KA, 256, 0, stream>>>(out);
}